// SVFeatureBlock_90125593739319
// MI455X (gfx1250) — compile-verified
//
#include <hip/hip_runtime.h>

typedef __attribute__((ext_vector_type(2))) float v2f;
typedef __attribute__((ext_vector_type(8))) float v8f;

#define BB 2048
#define CC 128
#define FF 512

// OR-reduce a value across each 16-lane half of a wave32 via ds_swizzle
// XOR butterflies (and=0x1F keeps lanes 0-15 / 16-31 groups separate).
__device__ __forceinline__ unsigned half16_or(unsigned v) {
    v |= (unsigned)__builtin_amdgcn_ds_swizzle((int)v, 0x041F); // xor 1
    v |= (unsigned)__builtin_amdgcn_ds_swizzle((int)v, 0x081F); // xor 2
    v |= (unsigned)__builtin_amdgcn_ds_swizzle((int)v, 0x101F); // xor 4
    v |= (unsigned)__builtin_amdgcn_ds_swizzle((int)v, 0x201F); // xor 8
    return v;
}

// One workgroup per sample b; 32 waves, wave w owns columns [16w, 16w+16).
// Column sums via v_wmma_f32_16x16x4_f32 with A = ones(16x4) (exact f32 math).
// Row validity kept branchless in per-lane register bitmasks, folded once
// after the loop (ds_swizzle OR + two LDS atomicOrs per half-wave).
__global__ __launch_bounds__(1024, 1)
void svfeat_mean_kernel(const float* __restrict__ x, float* __restrict__ out) {
    const int b    = blockIdx.x;
    const int tid  = threadIdx.x;
    const int lane = tid & 31;
    const int wv   = tid >> 5;          // 0..31 -> 16-column tile index
    const int n    = lane & 15;         // column within tile (N)
    const int kk   = (lane >> 4) << 1;  // row class 0 for lanes 0-15, 2 for 16-31

    __shared__ unsigned rowMask[4];     // word j: rows == j (mod 4); bit i -> row 4i+j
    if (tid < 4) rowMask[tid] = 0u;
    __syncthreads();

    const float* base =
        x + (size_t)b * (CC * FF) + (size_t)kk * FF + wv * 16 + n;

    const v2f aOnes = {1.0f, 1.0f};     // D[m,n] = sum_k B[k,n] + C[m,n]
    v8f acc = {};
    unsigned m0 = 0u, m1 = 0u;          // bit i: row 4i+kk (+1) nonzero at col n

#pragma unroll 8
    for (int i = 0; i < CC / 4; ++i) {
        // B(4x16) f32 fragment: VGPR0 = rows {4i, 4i+2}, VGPR1 = rows {4i+1, 4i+3}
        float b0 = __builtin_nontemporal_load(base + (size_t)(4 * i)     * FF);
        float b1 = __builtin_nontemporal_load(base + (size_t)(4 * i + 1) * FF);
        v2f bm = {b0, b1};

        acc = __builtin_amdgcn_wmma_f32_16x16x4_f32(
            /*neg_a=*/false, aOnes, /*neg_b=*/false, bm,
            /*c_mod=*/(short)0, acc, /*reuse_a=*/false, /*reuse_b=*/false);

        m0 |= (b0 != 0.0f) ? (1u << i) : 0u;
        m1 |= (b1 != 0.0f) ? (1u << i) : 0u;
    }

    // Fold validity: OR over the 16 columns of this tile per half-wave,
    // then one lane per half merges into the block-wide row bitmask.
    m0 = half16_or(m0);
    m1 = half16_or(m1);
    if ((lane & 15) == 0) {
        atomicOr(&rowMask[kk + 0], m0);
        atomicOr(&rowMask[kk + 1], m1);
    }
    __syncthreads();

    const float cnt = (float)(__popc(rowMask[0]) + __popc(rowMask[1]) +
                              __popc(rowMask[2]) + __popc(rowMask[3]));

    // All rows of D are identical (A all-ones); lanes 0-15 hold colsum[N] in acc[0].
    if (lane < 16) {
        out[(size_t)b * FF + wv * 16 + lane] = acc[0] / cnt;
    }
}

extern "C" void kernel_launch(void* const* d_in, const int* in_sizes, int n_in,
                              void* d_out, int out_size, void* d_ws, size_t ws_size,
                              hipStream_t stream) {
    const float* x = (const float*)d_in[0];
    float* out = (float*)d_out;
    (void)in_sizes; (void)n_in; (void)out_size; (void)d_ws; (void)ws_size;
    svfeat_mean_kernel<<<dim3(BB), dim3(1024), 0, stream>>>(x, out);
}